// GuidedFilterHR_fast_49443663511699
// MI455X (gfx1250) — compile-verified
//
#include <hip/hip_runtime.h>
#include <cmath>
#include <cstdint>

typedef float v2f __attribute__((ext_vector_type(2)));
typedef float v8f __attribute__((ext_vector_type(8)));

// [fam][tap]: fam 0 = db2 (L=4), fam 1 = db8 (L=16)
__constant__ float c_lo[2][16] = {
  {-0.12940952255092145f, 0.22414386804185735f, 0.836516303737469f, 0.48296291314469025f,
   0.f,0.f,0.f,0.f,0.f,0.f,0.f,0.f,0.f,0.f,0.f,0.f},
  {-0.00011747678400228192f, 0.0006754494059985568f, -0.0003917403729959771f, -0.00487035299301066f,
   0.008746094047015655f, 0.013981027917015516f, -0.04408825393106472f, -0.01736930100202211f,
   0.128747426620186f, 0.00047248457399797254f, -0.2840155429624281f, -0.015829105256023893f,
   0.5853546836548691f, 0.6756307362980128f, 0.3128715909144659f, 0.05441584224308161f}
};
__constant__ float c_hi[2][16] = {
  {0.48296291314469025f, -0.836516303737469f, 0.22414386804185735f, 0.12940952255092145f,
   0.f,0.f,0.f,0.f,0.f,0.f,0.f,0.f,0.f,0.f,0.f,0.f},
  {0.05441584224308161f, -0.3128715909144659f, 0.6756307362980128f, -0.5853546836548691f,
   -0.015829105256023893f, 0.2840155429624281f, 0.00047248457399797254f, -0.128747426620186f,
   -0.01736930100202211f, 0.04408825393106472f, 0.013981027917015516f, -0.008746094047015655f,
   -0.00487035299301066f, 0.0003917403729959771f, 0.0006754494059985568f, 0.00011747678400228192f}
};

// Branchless single-fold reflections. Validity of a single fold is proven per
// call-site: overshoot never exceeds one reflection period.
// numpy/jnp.pad 'reflect' (mirror, excludes edge): -1 -> 1, n -> n-2
__device__ __forceinline__ int reflect_ex1(int i, int n) {
  i = (i < 0) ? -i : i;
  i = (i >= n) ? (2 * n - 2 - i) : i;
  return i;
}
// scipy/jax map_coordinates 'reflect' (symmetric, includes edge): -1 -> 0, n -> n-1
__device__ __forceinline__ int reflect_sym1(int i, int n) {
  i = (i < 0) ? (-i - 1) : i;
  i = (i >= n) ? (2 * n - 1 - i) : i;
  return i;
}

// ---------------- bilinear warp (map_coordinates order=1, mode='reflect') ---------
__global__ void k_warp(const float* __restrict__ img, const float* __restrict__ cy,
                       const float* __restrict__ cx, float* __restrict__ out, int H, int W) {
  int idx = blockIdx.x * blockDim.x + threadIdx.x;
  if (idx >= H * W) return;
  float y = cy[idx], x = cx[idx];
  float fy = floorf(y), fx = floorf(x);
  int y0 = (int)fy, x0 = (int)fx;
  float wy = y - fy, wx = x - fx;
  int y0r = reflect_sym1(y0, H), y1r = reflect_sym1(y0 + 1, H);
  int x0r = reflect_sym1(x0, W), x1r = reflect_sym1(x0 + 1, W);
  float v00 = img[y0r * W + x0r], v01 = img[y0r * W + x1r];
  float v10 = img[y1r * W + x0r], v11 = img[y1r * W + x1r];
  out[idx] = v00*(1.f-wy)*(1.f-wx) + v01*(1.f-wy)*wx + v10*wy*(1.f-wx) + v11*wy*wx;
}

// ---------------- forward DWT along one axis (WMMA f32 16x16x4 banded matmul) -----
// Tile: M = 16 lines (batch), N = 16 output positions.  C = sum_j A_j(16x4) B_j(4x16)
// A_j[m][k] = src[line m][reflect_ex(2*n0 - padl + 4j + k)]
// B_j[k][n] = f[4j + k - 2n]  (0 <= tap < L)
__global__ void __launch_bounds__(32)
k_dwt(const float* __restrict__ src, int R, int lsIn, int asIn, int n_in, int n_out, int fam,
      float* __restrict__ outA, float* __restrict__ outD, int lsOut, int asOut) {
  const int L = fam ? 16 : 4;
  const int padl = L - 2;
  const float* lo = c_lo[fam];
  const float* hi = c_hi[fam];
  int lane = threadIdx.x;
  int m0 = blockIdx.y << 4;
  int n0 = blockIdx.x << 4;
  int ml   = lane & 15;      // A: row M / B: col N / C: col N
  int half = lane >> 4;
  int kb   = half << 1;      // K base for this half-wave
#if defined(__HIP_DEVICE_COMPILE__) && __has_builtin(__builtin_amdgcn_wmma_f32_16x16x4_f32)
  const int J = (L + 33) >> 2;   // 9 chunks for db2, 12 for db8
  int line = m0 + ml; if (line >= R) line = R - 1;   // clamp (keeps EXEC uniform)
  int lbase = line * lsIn;
  int qbase = 2 * n0 - padl;
  int nfold = 2 * n_in - 2;
  int nb2 = ml << 1;
  v8f cA = {}; v8f cD = {};
  for (int j = 0; j < J; ++j) {
    int pos = (j << 2) + kb;
    int q0 = qbase + pos;
    int q1 = q0 + 1;
    q0 = (q0 < 0) ? -q0 : q0;  q0 = (q0 >= n_in) ? (nfold - q0) : q0;
    q1 = (q1 < 0) ? -q1 : q1;  q1 = (q1 >= n_in) ? (nfold - q1) : q1;
    v2f av;
    av.x = src[lbase + q0 * asIn];
    av.y = src[lbase + q1 * asIn];
    int t0 = pos - nb2;
    bool in0 = (unsigned)t0       < (unsigned)L;
    bool in1 = (unsigned)(t0 + 1) < (unsigned)L;
    float l0 = lo[t0 & 15], l1 = lo[(t0 + 1) & 15];
    float h0 = hi[t0 & 15], h1 = hi[(t0 + 1) & 15];
    v2f bl, bh;
    bl.x = in0 ? l0 : 0.f;  bl.y = in1 ? l1 : 0.f;
    bh.x = in0 ? h0 : 0.f;  bh.y = in1 ? h1 : 0.f;
    cA = __builtin_amdgcn_wmma_f32_16x16x4_f32(false, av, false, bl, (short)0, cA, false, false);
    cD = __builtin_amdgcn_wmma_f32_16x16x4_f32(false, av, false, bh, (short)0, cD, false, false);
  }
  int nIdx = n0 + ml;
  int mh = half << 3;
  for (int r = 0; r < 8; ++r) {
    int lr = m0 + mh + r;
    if (lr < R && nIdx < n_out) {
      int oo = lr * lsOut + nIdx * asOut;
      outA[oo] = cA[r];
      outD[oo] = cD[r];
    }
  }
#else
  // scalar fallback (same geometry, identical mapping)
  int nIdx = n0 + ml;
  int mh = half << 3;
  int nfold = 2 * n_in - 2;
  for (int r = 0; r < 8; ++r) {
    int lr = m0 + mh + r;
    if (lr < R && nIdx < n_out) {
      float sa = 0.f, sd = 0.f;
      for (int k = 0; k < L; ++k) {
        int q = 2 * nIdx + k - padl;
        q = (q < 0) ? -q : q;  q = (q >= n_in) ? (nfold - q) : q;
        float v = src[lr * lsIn + q * asIn];
        sa += v * lo[k]; sd += v * hi[k];
      }
      int oo = lr * lsOut + nIdx * asOut;
      outA[oo] = sa; outD[oo] = sd;
    }
  }
#endif
}

// ---------------- inverse DWT along one axis:  y[m] = sum_i a[i] lo[m+padl-2i] + d[i] hi[...]
__global__ void k_idwt(const float* __restrict__ a, int aLs, int aAs,
                       const float* __restrict__ d, int dLs, int dAs,
                       int nc, int fam, int n, int R,
                       float* __restrict__ y, int yLs, int yAs) {
  int idx = blockIdx.x * blockDim.x + threadIdx.x;
  if (idx >= R * n) return;
  int line = idx / n, m = idx % n;
  const int L = fam ? 16 : 4;
  const int padl = L - 2;
  const float* lo = c_lo[fam];
  const float* hi = c_hi[fam];
  int t = m + padl;
  int raw = t - L + 1;
  int i0 = raw > 0 ? (raw + 1) >> 1 : 0;
  int i1 = t >> 1; if (i1 > nc - 1) i1 = nc - 1;
  float acc = 0.f;
  int ab = line * aLs, db = line * dLs;
  for (int i = i0; i <= i1; ++i) {
    int tap = t - 2 * i;
    acc += a[ab + i * aAs] * lo[tap] + d[db + i * dAs] * hi[tap];
  }
  y[line * yLs + m * yAs] = acc;
}

// ---------------- abs-comparison detail injection: where(|a| ? |b|, a, b) ----------
__global__ void k_select(float* dst, int dLs, const float* a, const float* b, int sLs,
                         int h, int w, int larger) {
  int idx = blockIdx.x * blockDim.x + threadIdx.x;
  if (idx >= h * w) return;
  int r = idx / w, c = idx % w;
  float va = a[r * sLs + c], vb = b[r * sLs + c];
  bool take_a = larger ? (fabsf(va) > fabsf(vb)) : (fabsf(va) < fabsf(vb));
  dst[r * dLs + c] = take_a ? va : vb;
}

// ---------------- pad-by-1 reflect (numpy 'reflect': new edge = src[n-2]) ----------
__global__ void k_pad1(const float* __restrict__ src, int hs, int ws, int sLs,
                       float* __restrict__ dst, int dLs) {
  int hd = hs + 1, wd = ws + 1;
  int idx = blockIdx.x * blockDim.x + threadIdx.x;
  if (idx >= hd * wd) return;
  int i = idx / wd, j = idx % wd;
  int si = (i == hs) ? hs - 2 : i;
  int sj = (j == ws) ? ws - 2 : j;
  dst[i * dLs + j] = src[si * sLs + sj];
}

__global__ void k_sub(float* dst, const float* a, const float* b, int n) {
  int i = blockIdx.x * blockDim.x + threadIdx.x;
  if (i < n) dst[i] = a[i] - b[i];
}
__global__ void k_add(float* dst, const float* a, const float* b, int n) {
  int i = blockIdx.x * blockDim.x + threadIdx.x;
  if (i < n) dst[i] = a[i] + b[i];
}

// ---------------- 89-tap vertical sliding median (exact 44th order statistic) ------
#define MW 89
#define MR 44
__global__ void __launch_bounds__(256) k_median(const float* __restrict__ src,
                                                float* __restrict__ dst, int H, int W) {
  __shared__ unsigned keys[96][33];
  int c0 = blockIdx.x * 32;
  int r0 = blockIdx.y * 8;
  int tid = threadIdx.x;
  for (int t = tid; t < 96 * 32; t += 256) {
    int rr = t >> 5, cc = t & 31;
    int gr = reflect_ex1(r0 - MR + rr, H);
    int gc = c0 + cc; if (gc >= W) gc = W - 1;
    unsigned u = __float_as_uint(src[gr * W + gc]);
    keys[rr][cc] = (u & 0x80000000u) ? ~u : (u | 0x80000000u);  // order-preserving key
  }
  __syncthreads();
  int lr = tid >> 5, lc = tid & 31;
  unsigned P = 0u; int need = MR;
  for (int b = 31; b >= 0; --b) {
    unsigned maskAbove = (b < 31) ? (0xFFFFFFFFu << (b + 1)) : 0u;
    int cnt = 0;
    for (int t = 0; t < MW; ++t) {
      unsigned k = keys[lr + t][lc];
      if ((k & maskAbove) == P && !((k >> b) & 1u)) cnt++;
    }
    if (need >= cnt) { need -= cnt; P |= (1u << b); }
  }
  int orow = r0 + lr, ocol = c0 + lc;
  if (orow < H && ocol < W) {
    unsigned u = (P & 0x80000000u) ? (P ^ 0x80000000u) : ~P;
    dst[orow * W + ocol] = __uint_as_float(u);
  }
}

// =============================== host orchestration ================================
static inline int ceil16(int x) { return (x + 15) >> 4; }

static void launch_dwt(const float* src, int R, int lsIn, int asIn, int n_in, int n_out,
                       int fam, float* oA, float* oD, int lsOut, int asOut, hipStream_t st) {
  dim3 g(ceil16(n_out), ceil16(R));
  k_dwt<<<g, 32, 0, st>>>(src, R, lsIn, asIn, n_in, n_out, fam, oA, oD, lsOut, asOut);
}
static void launch_idwt(const float* a, int aLs, int aAs, const float* d, int dLs, int dAs,
                        int nc, int fam, int n, int R, float* y, int yLs, int yAs, hipStream_t st) {
  int total = R * n;
  k_idwt<<<(total + 255) / 256, 256, 0, st>>>(a, aLs, aAs, d, dLs, dAs, nc, fam, n, R, y, yLs, yAs);
}

struct Scr { float *ROW0, *ROW1, *T[6], *P0, *P1, *Q0, *Q1, *MD; };

// wave_rec: level-6 wavedec2 of y and X (base == y), detail injection, waverec2.
static void run_waverec(const float* yimg, int ys, const float* Ximg, int xs, int dim,
                        int fam, int larger, float* outF, int outS, const Scr& sc, hipStream_t st) {
  const int L = fam ? 16 : 4;
  int nd[7]; nd[0] = dim;
  for (int l = 0; l < 6; ++l) nd[l + 1] = (nd[l] + L - 1) >> 1;
  long mdOff[6]; long mo = 0;
  for (int l = 0; l < 6; ++l) { mdOff[l] = mo; mo += 3L * nd[l + 1] * nd[l + 1]; }

  const float *cy = yimg, *cx = Ximg; int cys = ys, cxs = xs;
  float* pp[2] = { sc.P0, sc.P1 };
  float* qq[2] = { sc.Q0, sc.Q1 };
  for (int l = 0; l < 6; ++l) {
    int h = nd[l], na = nd[l + 1];
    float* Pn = pp[l & 1]; float* Qn = qq[l & 1];
    // Y: rows then cols (lw -> ll,lh ; hw -> hl,hh)
    launch_dwt(cy, h, cys, 1, h, na, fam, sc.ROW0, sc.ROW1, 520, 1, st);
    launch_dwt(sc.ROW0, na, 1, 520, h, na, fam, Pn,       sc.T[0], 1, 520, st);
    launch_dwt(sc.ROW1, na, 1, 520, h, na, fam, sc.T[1],  sc.T[2], 1, 520, st);
    // X
    launch_dwt(cx, h, cxs, 1, h, na, fam, sc.ROW0, sc.ROW1, 520, 1, st);
    launch_dwt(sc.ROW0, na, 1, 520, h, na, fam, Qn,       sc.T[3], 1, 520, st);
    launch_dwt(sc.ROW1, na, 1, 520, h, na, fam, sc.T[4],  sc.T[5], 1, 520, st);
    // merged details (packed, stride na)
    float* md = sc.MD + mdOff[l];
    long nn = (long)na * na; int blocks = (int)((nn + 255) / 256);
    k_select<<<blocks, 256, 0, st>>>(md,          na, sc.T[0], sc.T[3], 520, na, na, larger);
    k_select<<<blocks, 256, 0, st>>>(md + nn,     na, sc.T[1], sc.T[4], 520, na, na, larger);
    k_select<<<blocks, 256, 0, st>>>(md + 2 * nn, na, sc.T[2], sc.T[5], 520, na, na, larger);
    cy = Pn; cys = 520; cx = Qn; cxs = 520;
  }
  // reconstruction (approx = y's LL6, stride 520; Q buffers as ping-pong)
  const float* cur = cy; int curS = 520;
  int tog = 0;
  for (int l = 5; l >= 0; --l) {
    int h = nd[l], na = nd[l + 1];
    const float* md = sc.MD + mdOff[l];
    long nn = (long)na * na;
    launch_idwt(cur,      1, curS, md,          1, na, na, fam, h, na, sc.ROW0, 1, 520, st); // LW
    launch_idwt(md + nn,  1, na,   md + 2 * nn, 1, na, na, fam, h, na, sc.ROW1, 1, 520, st); // HW
    float* dst; int dstS;
    if (l == 0) { dst = outF; dstS = outS; } else { dst = qq[tog]; dstS = 520; tog ^= 1; }
    launch_idwt(sc.ROW0, 520, 1, sc.ROW1, 520, 1, na, fam, h, h, dst, dstS, 1, st);
    cur = dst; curS = dstS;
  }
}

extern "C" void kernel_launch(void* const* d_in, const int* in_sizes, int n_in,
                              void* d_out, int out_size, void* d_ws, size_t ws_size,
                              hipStream_t stream) {
  (void)in_sizes; (void)n_in; (void)out_size; (void)ws_size;
  const float* xx   = (const float*)d_in[0];
  const float* yy   = (const float*)d_in[1];
  const float* coor = (const float*)d_in[2];
  const float* hX   = (const float*)d_in[3];
  float* out = (float*)d_out;
  const int H = 1024, W = 1024;
  const int HP = 1025;

  float* base = (float*)d_ws;
  size_t off = 0;
  auto take = [&](size_t n) -> float* { float* p = base + off; off += (n + 63) & ~(size_t)63; return p; };
  float* RECON = take((size_t)H * W);
  float* HXW   = take((size_t)H * W);
  float* R1023 = take(1023u * 1023u);
  float* TP    = take((size_t)H * W);
  float* Y1    = take((size_t)HP * HP);
  float* X1    = take((size_t)HP * HP);
  float* HXB   = take((size_t)HP * HP);
  float* BMED  = take((size_t)HP * HP);
  float* RES   = take((size_t)HP * HP);
  Scr sc;
  sc.ROW0 = take(1024u * 520u);
  sc.ROW1 = take(1024u * 520u);
  for (int i = 0; i < 6; ++i) sc.T[i] = take(520u * 520u);
  sc.P0 = take(520u * 520u); sc.P1 = take(520u * 520u);
  sc.Q0 = take(520u * 520u); sc.Q1 = take(520u * 520u);
  sc.MD = take(1250000u);

  int npix = H * W, eb = (npix + 255) / 256;
  // 1) warps: hXw = warp(xx), recon = warp(yy)
  k_warp<<<eb, 256, 0, stream>>>(xx, coor, coor + (size_t)H * W, HXW,   H, W);
  k_warp<<<eb, 256, 0, stream>>>(yy, coor, coor + (size_t)H * W, RECON, H, W);

  // 2) db2 wave_rec on recon (crop 1023x1023), then reflect-pad x2 -> Y1 (1025^2)
  run_waverec(RECON, W, hX, W, 1023, /*fam=*/0, /*larger=*/0, R1023, 1023, sc, stream);
  k_pad1<<<(1024 * 1024 + 255) / 256, 256, 0, stream>>>(R1023, 1023, 1023, 1023, TP, 1024);
  k_pad1<<<(HP * HP + 255) / 256, 256, 0, stream>>>(TP, 1024, 1024, 1024, Y1, HP);

  // 3) db2 wave_rec on hXw -> X1 (1025^2)
  run_waverec(HXW, W, hX, W, 1023, 0, 0, R1023, 1023, sc, stream);
  k_pad1<<<(1024 * 1024 + 255) / 256, 256, 0, stream>>>(R1023, 1023, 1023, 1023, TP, 1024);
  k_pad1<<<(HP * HP + 255) / 256, 256, 0, stream>>>(TP, 1024, 1024, 1024, X1, HP);

  // 4) hx1 = pad(hX) -> HXB (1025^2)
  k_pad1<<<(HP * HP + 255) / 256, 256, 0, stream>>>(hX, 1024, 1024, 1024, HXB, HP);

  // 5) b = sliding_median_v(Y1 - X1); HXB += b
  int np = HP * HP, ebp = (np + 255) / 256;
  k_sub<<<ebp, 256, 0, stream>>>(RES, Y1, X1, np);
  dim3 mg((HP + 31) / 32, (HP + 7) / 8);
  k_median<<<mg, 256, 0, stream>>>(RES, BMED, HP, HP);
  k_add<<<ebp, 256, 0, stream>>>(HXB, HXB, BMED, np);

  // 6) db8 wave_rec(y1, hx1+b, y1, larger=True); final pad+crop cancel -> d_out
  run_waverec(Y1, HP, HXB, HP, 1024, /*fam=*/1, /*larger=*/1, out, W, sc, stream);
}